// KAN_45114336477502
// MI455X (gfx1250) — compile-verified
//
#include <hip/hip_runtime.h>
#include <hip/hip_bf16.h>

typedef __attribute__((ext_vector_type(2))) float v2f;
typedef __attribute__((ext_vector_type(8))) float v8f;

#define QSP_DEPTH 10
#define NPH   (2 * QSP_DEPTH + 1)   // 21 phase params
#define NCOEF (QSP_DEPTH + 1)       // 11 Chebyshev coefficients (even harmonics 0..20)

// DCT-I inversion matrix: beta_j = sum_k C[j][k] * g_k,
// C[j][k] = (2/N) * eps_j * eps_k * cos(pi*j*k/N), N = QSP_DEPTH, eps_{0,N} = 1/2.
// j*k is reduced mod 2N exactly in integer math so the native cos argument is <= 2*pi.
// All scaling factors are compile-time constants (no runtime fp division).
__device__ __forceinline__ float dct_c(int j, int k) {
    if (j >= NCOEF || k >= NCOEF) return 0.0f;
    const float PI_OVER_N = 3.14159265358979323846f / (float)QSP_DEPTH;
    const float TWO_OVER_N = 2.0f / (float)QSP_DEPTH;
    float ej = (j == 0 || j == QSP_DEPTH) ? 0.5f : 1.0f;
    float ek = (k == 0 || k == QSP_DEPTH) ? 0.5f : 1.0f;
    int m = (j * k) % (2 * QSP_DEPTH);                       // exact periodic reduction
    return TWO_OVER_N * ej * ek * __cosf(PI_OVER_N * (float)m);
}

// ---------------------------------------------------------------------------
// Kernel 1: one wave. Lanes 0..10 evaluate the exact QSP recurrence at the
// DCT-I nodes theta_k = pi*k/40, then a WMMA 16x16x16 (4 x 16x16x4 f32 chain)
// applies the DCT-I matrix to recover the 11 Chebyshev coefficients beta.
// All trig uses native v_cos_f32/v_sin_f32 (args are in [0, 2*pi)).
// ---------------------------------------------------------------------------
__global__ __launch_bounds__(32)
void qsp_coeff_kernel(const float* __restrict__ qsp, float* __restrict__ beta_out) {
    const int lane = threadIdx.x;
    const float PI = 3.14159265358979323846f;

    float ph[NPH];
#pragma unroll
    for (int t = 0; t < NPH; ++t) ph[t] = qsp[t];

    // Node for this lane (lanes >= 11 compute dummy values, masked below)
    float theta = (PI / (4.0f * (float)QSP_DEPTH)) * (float)lane; // pi*k/40
    float c = __cosf(theta);
    float s = __sinf(theta);

    // Row 0 of M: u0 = e^{i phi0}, u1 = 0
    float a = __cosf(ph[0]);   // Re u0
    float b = __sinf(ph[0]);   // Im u0
    float d = 0.0f;            // Re u1
    float e = 0.0f;            // Im u1

#pragma unroll
    for (int t = 1; t < NPH; ++t) {
        // u' = u * W,  W = [[c, i s],[i s, c]]
        float ar = a * c - e * s;
        float ai = b * c + d * s;
        float dr = d * c - b * s;
        float di = e * c + a * s;
        // column phases: u0 *= e^{+i phi}, u1 *= e^{-i phi}
        float cp = __cosf(ph[t]);
        float sp = __sinf(ph[t]);
        a = ar * cp - ai * sp;
        b = ar * sp + ai * cp;
        d = dr * cp + di * sp;
        e = di * cp - dr * sp;
    }

    float g = (lane < NCOEF) ? a : 0.0f;   // g_k = Re(U(theta_k)[0,0])

    __shared__ float gs[32];
    gs[lane] = g;
    __syncthreads();

    // WMMA: D = A x B, A[m][k] = g[K] replicated over rows m (K = 4*ch + k),
    // B[k][n] = C[n][K]. Every row of D equals beta; D VGPR0 lanes 0..15 = beta_n.
    const bool hi = (lane >= 16);
    const int  n  = lane & 15;

    v8f acc = {};
#pragma unroll
    for (int ch = 0; ch < 4; ++ch) {
        // A 16x4 f32 layout: VGPR0 <- K=0 (lanes 0-15) / K=2 (lanes 16-31); VGPR1 <- K=1 / K=3
        v2f A;
        A.x = gs[4 * ch + (hi ? 2 : 0)];
        A.y = gs[4 * ch + (hi ? 3 : 1)];
        // B 4x16 f32 layout (row-striped): N = lane%16, K = vgpr + 2*(lane>=16)
        v2f B;
        B.x = dct_c(n, 4 * ch + (hi ? 2 : 0));
        B.y = dct_c(n, 4 * ch + (hi ? 3 : 1));
        acc = __builtin_amdgcn_wmma_f32_16x16x4_f32(
            /*neg_a=*/false, A, /*neg_b=*/false, B,
            /*c_mod=*/(short)0, acc, /*reuse_a=*/false, /*reuse_b=*/false);
    }

    if (lane < 16) beta_out[lane] = acc[0];  // beta_11..15 are exact zeros
}

// ---------------------------------------------------------------------------
// Kernel 2: memory-bound streaming evaluation.
// out[i] = alpha[i] * sum_{j=0..10} beta_j * T_j(cos(2*x[i]))
// 12 B/sample of HBM traffic, ~20 VALU ops/sample -> HBM-roofline bound.
// ---------------------------------------------------------------------------
__device__ __forceinline__ float cheb_eval(float x, const float* __restrict__ b) {
    float t  = __cosf(x + x);          // cos(2x)
    float t2 = t + t;
    float tm = 1.0f;
    float tc = t;
    float acc = fmaf(b[1], t, b[0]);
#pragma unroll
    for (int j = 2; j <= QSP_DEPTH; ++j) {
        float tn = fmaf(t2, tc, -tm);  // T_j = 2 t T_{j-1} - T_{j-2}
        acc = fmaf(b[j], tn, acc);
        tm = tc;
        tc = tn;
    }
    return acc;
}

__global__ __launch_bounds__(256)
void qsp_eval_kernel(const float* __restrict__ x,
                     const float* __restrict__ alphas,
                     const float* __restrict__ beta,
                     float* __restrict__ out, int n) {
    // beta: 12 floats (beta[11] == 0), loaded as three float4s; uniform, L0-resident
    float b[12];
    float4 b0 = ((const float4*)beta)[0];
    float4 b1 = ((const float4*)beta)[1];
    float4 b2 = ((const float4*)beta)[2];
    b[0] = b0.x; b[1] = b0.y; b[2]  = b0.z; b[3]  = b0.w;
    b[4] = b1.x; b[5] = b1.y; b[6]  = b1.z; b[7]  = b1.w;
    b[8] = b2.x; b[9] = b2.y; b[10] = b2.z; b[11] = b2.w;

    const int i4 = blockIdx.x * blockDim.x + threadIdx.x;
    const int n4 = n >> 2;

    if (i4 < n4) {
        float4 xv = ((const float4*)x)[i4];
        float4 av = ((const float4*)alphas)[i4];
        float4 ov;
        ov.x = av.x * cheb_eval(xv.x, b);
        ov.y = av.y * cheb_eval(xv.y, b);
        ov.z = av.z * cheb_eval(xv.z, b);
        ov.w = av.w * cheb_eval(xv.w, b);
        ((float4*)out)[i4] = ov;
    }
    // scalar tail (N % 4 != 0); never taken for N = 4M but kept general
    const int tail = n - (n4 << 2);
    if (i4 < tail) {
        int i = (n4 << 2) + i4;
        out[i] = alphas[i] * cheb_eval(x[i], b);
    }
}

// ---------------------------------------------------------------------------
extern "C" void kernel_launch(void* const* d_in, const int* in_sizes, int n_in,
                              void* d_out, int out_size, void* d_ws, size_t ws_size,
                              hipStream_t stream) {
    const float* x      = (const float*)d_in[0];
    const float* qsp    = (const float*)d_in[1];
    const float* alphas = (const float*)d_in[2];
    float*       out    = (float*)d_out;
    float*       beta   = (float*)d_ws;          // 16 floats of scratch
    const int n = in_sizes[0];

    qsp_coeff_kernel<<<1, 32, 0, stream>>>(qsp, beta);

    int n4 = n >> 2;
    int blocks = (n4 + 255) / 256;
    if (blocks < 1) blocks = 1;
    qsp_eval_kernel<<<blocks, 256, 0, stream>>>(x, alphas, beta, out, n);
}